// ExpandingAttention_35845797052447
// MI455X (gfx1250) — compile-verified
//
#include <hip/hip_runtime.h>
#include <hip/hip_bf16.h>
#include <math.h>

#define T_TOK 16384
#define C_EMB 1024
#define N_ITERS 30
#define ATT_SCALE (0.001f / 32.0f) // 0.001/sqrt(1024)

typedef float v2f __attribute__((ext_vector_type(2)));
typedef float v4f __attribute__((ext_vector_type(4)));
typedef float v8f __attribute__((ext_vector_type(8)));

// ---------------------------------------------------------------------------
// Wave (32-lane) reductions
// ---------------------------------------------------------------------------
__device__ __forceinline__ float wave_sum(float v) {
    #pragma unroll
    for (int o = 16; o > 0; o >>= 1) v += __shfl_xor(v, o, 32);
    return v;
}
__device__ __forceinline__ float wave_max(float v) {
    #pragma unroll
    for (int o = 16; o > 0; o >>= 1) v = fmaxf(v, __shfl_xor(v, o, 32));
    return v;
}

// ---------------------------------------------------------------------------
// Kernel A1: q_last[j] = sum_c x[T-1,c] * W[j,c]   (one wave per j)
// grid 128 x 256 threads (8 waves/block) -> 1024 waves
// ---------------------------------------------------------------------------
__global__ __launch_bounds__(256) void qlast_kernel(const float* __restrict__ x,
                                                    const float* __restrict__ W,
                                                    float* __restrict__ qlast) {
    const float* xl = x + (size_t)(T_TOK - 1) * C_EMB;
    int wid  = threadIdx.x >> 5;
    int lane = threadIdx.x & 31;
    int j = blockIdx.x * 8 + wid;
    const float* wr = W + (size_t)j * C_EMB;
    float s = 0.f;
    #pragma unroll 4
    for (int c = lane; c < C_EMB; c += 32) s = fmaf(xl[c], wr[c], s);
    s = wave_sum(s);
    if (lane == 0) qlast[j] = s;
}

// ---------------------------------------------------------------------------
// Kernel A2a: partial[by][c] = sum_{j in 128-chunk} qlast[j] * W[1024+j, c]
// grid (4, 8) x 256 : coalesced over c, deterministic (no atomics)
// ---------------------------------------------------------------------------
__global__ __launch_bounds__(256) void u_partial_kernel(const float* __restrict__ W,
                                                        const float* __restrict__ qlast,
                                                        float* __restrict__ part) {
    int c  = blockIdx.x * 256 + threadIdx.x;
    int j0 = blockIdx.y * 128;
    float s = 0.f;
    #pragma unroll 4
    for (int j = j0; j < j0 + 128; ++j)
        s = fmaf(qlast[j], W[(size_t)(C_EMB + j) * C_EMB + c], s);
    part[(size_t)blockIdx.y * C_EMB + c] = s;
}

// Kernel A2b: u[c] = sum_by partial[by][c]
__global__ __launch_bounds__(256) void u_reduce_kernel(const float* __restrict__ part,
                                                       float* __restrict__ u) {
    int c = blockIdx.x * 256 + threadIdx.x;
    float s = 0.f;
    #pragma unroll
    for (int by = 0; by < 8; ++by) s += part[(size_t)by * C_EMB + c];
    u[c] = s;
}

// ---------------------------------------------------------------------------
// Kernel B: fused bandwidth pass over x using V_WMMA_F32_16X16X4_F32.
//   att[t] = scale * (u . x[t]),  v[t] = (w_v . x[t])
// One wave per 16-row tile of x. B-matrix: col0 = u chunk, col1 = wv chunk,
// cols 2..15 zero. LDS holds THREE 1024-float rows: u, wv, and a zero row;
// lane n selects its row with sel = min(n,2) so every lane issues an
// unconditional ds_load_b128 (no EXEC juggling, shared addresses broadcast).
// K processed 8 at a time with TWO independent accumulator chains (c0,c1).
// x loads are non-temporal (x is streamed exactly once = entire roofline).
//
// K-slot -> column mapping (per outer step i, cols [8i, 8i+8)):
//   WMMA#1: slots(0,1)=cols 8i+0,8i+1 (lanes 0-15), slots(2,3)=cols 8i+4,8i+5
//   WMMA#2: slots(0,1)=cols 8i+2,8i+3,             slots(2,3)=cols 8i+6,8i+7
// so lane l (half h=l>>4) loads float4 at col 8i+4h feeding both WMMAs.
// ---------------------------------------------------------------------------
__global__ __launch_bounds__(256) void fused_att_v_kernel(const float* __restrict__ x,
                                                          const float* __restrict__ u,
                                                          const float* __restrict__ wv,
                                                          float* __restrict__ att,
                                                          float* __restrict__ v) {
    __shared__ float sb[3 * C_EMB];   // [0]=u, [1]=wv, [2]=zeros
    for (int i = threadIdx.x; i < C_EMB; i += 256) {
        sb[i]              = u[i];
        sb[C_EMB + i]      = wv[i];
        sb[2 * C_EMB + i]  = 0.f;
    }
    __syncthreads();

    const int wid  = threadIdx.x >> 5;
    const int lane = threadIdx.x & 31;
    const int n = lane & 15;   // N (column of C/D); also M (x row in tile) for A
    const int h = lane >> 4;   // half select: K-slot pair

    const int tile = blockIdx.x * 8 + wid;   // 0..1023
    const int row0 = tile * 16;
    const float* xrow = x + (size_t)(row0 + n) * C_EMB;
    const int sel = (n < 2) ? n : 2;
    const float* bp = &sb[sel * C_EMB];

    v8f c0 = {0.f, 0.f, 0.f, 0.f, 0.f, 0.f, 0.f, 0.f};
    v8f c1 = {0.f, 0.f, 0.f, 0.f, 0.f, 0.f, 0.f, 0.f};

    #pragma unroll 8
    for (int i = 0; i < C_EMB / 8; ++i) {
        const int col = 8 * i + 4 * h;
        v4f xa = __builtin_nontemporal_load((const v4f*)(xrow + col));
        v4f bb = *(const v4f*)(bp + col);
        v2f a1 = {xa.x, xa.y};
        v2f a2 = {xa.z, xa.w};
        v2f b1 = {bb.x, bb.y};
        v2f b2 = {bb.z, bb.w};
        c0 = __builtin_amdgcn_wmma_f32_16x16x4_f32(false, a1, false, b1,
                                                   (short)0, c0, false, false);
        c1 = __builtin_amdgcn_wmma_f32_16x16x4_f32(false, a2, false, b2,
                                                   (short)0, c1, false, false);
    }

    v8f d = c0 + c1;
    // D layout: lane<16 -> N=lane, M=r ; lane>=16 -> N=lane-16, M=8+r
    if (n == 0) {
        #pragma unroll
        for (int r = 0; r < 8; ++r) att[row0 + 8 * h + r] = d[r] * ATT_SCALE;
    } else if (n == 1) {
        #pragma unroll
        for (int r = 0; r < 8; ++r) v[row0 + 8 * h + r] = d[r];
    }
}

// ---------------------------------------------------------------------------
// Kernel C: 30-iteration expanding-window softmax scan + final y. One block.
// p never materialized: y = Sv/Z, bu = Sc/Z with shared-max-stabilized exp.
// ---------------------------------------------------------------------------
__device__ __forceinline__ float block_max(float val, float* red) {
    int wid = threadIdx.x >> 5, lane = threadIdx.x & 31;
    val = wave_max(val);
    if (lane == 0) red[wid] = val;
    __syncthreads();
    if (wid == 0) {
        float r = red[lane];
        r = wave_max(r);
        if (lane == 0) red[0] = r;
    }
    __syncthreads();
    float out = red[0];
    __syncthreads();
    return out;
}

__device__ __forceinline__ void block_sum3(float& a, float& b, float& c, float* red) {
    int wid = threadIdx.x >> 5, lane = threadIdx.x & 31;
    a = wave_sum(a); b = wave_sum(b); c = wave_sum(c);
    if (lane == 0) { red[wid] = a; red[32 + wid] = b; red[64 + wid] = c; }
    __syncthreads();
    if (wid == 0) {
        float ra = wave_sum(red[lane]);
        float rb = wave_sum(red[32 + lane]);
        float rc = wave_sum(red[64 + lane]);
        if (lane == 0) { red[0] = ra; red[32] = rb; red[64] = rc; }
    }
    __syncthreads();
    a = red[0]; b = red[32]; c = red[64];
    __syncthreads();
}

__global__ __launch_bounds__(1024) void scan_kernel(const float* __restrict__ att,
                                                    const float* __restrict__ v,
                                                    const float* __restrict__ alphap,
                                                    const float* __restrict__ betap,
                                                    float* __restrict__ out) {
    __shared__ float red[96];
    const int tid = threadIdx.x;
    float a = alphap[0];
    float b = betap[0];
    float k_old = 0.f;
    float y = 0.f;

    for (int it = 0; it < N_ITERS; ++it) {
        float kk = 2.f * (a + b) / a;
        float window = ceilf(kk);
        float wclamp = fminf(window, (float)T_TOK);
        int start = T_TOK - (int)wclamp;
        if (start < 0) start = 0;

        // t == T-1 is forced to -inf in the reference -> exclude (contrib 0)
        float m = -INFINITY;
        for (int t = start + tid; t < T_TOK - 1; t += 1024)
            m = fmaxf(m, att[t]);
        m = block_max(m, red);

        float Z = 0.f, Sc = 0.f, Sv = 0.f;
        for (int t = start + tid; t < T_TOK - 1; t += 1024) {
            float e = expf(att[t] - m);
            Z += e;
            Sc = fmaf(e, (float)(T_TOK - 1 - t), Sc);
            Sv = fmaf(e, v[t], Sv);
        }
        block_sum3(Z, Sc, Sv, red);

        float bu = Sc / Z;
        y = Sv / Z;                 // p (and thus y) updated this iteration
        a += 1.f;
        b += bu;
        bool done_now = (kk > (float)T_TOK) || (kk < k_old);
        k_old = kk;
        if (done_now) break;        // uniform across the block
    }
    if (tid == 0) out[0] = y;
}

// ---------------------------------------------------------------------------
// Launch
// ---------------------------------------------------------------------------
extern "C" void kernel_launch(void* const* d_in, const int* in_sizes, int n_in,
                              void* d_out, int out_size, void* d_ws, size_t ws_size,
                              hipStream_t stream) {
    const float* x     = (const float*)d_in[0];   // (1, 16384, 1024)
    const float* W     = (const float*)d_in[1];   // (2049, 1024)
    const float* alpha = (const float*)d_in[2];
    const float* beta  = (const float*)d_in[3];
    float* out = (float*)d_out;

    float* ws    = (float*)d_ws;
    float* qlast = ws;                       // 1024
    float* u     = ws + 1024;                // 1024
    float* part  = ws + 2048;                // 8 * 1024
    float* att   = ws + 2048 + 8 * 1024;     // 16384
    float* vv    = att + T_TOK;              // 16384

    const float* wv = W + (size_t)(2 * C_EMB) * C_EMB;  // W row 2048

    qlast_kernel<<<128, 256, 0, stream>>>(x, W, qlast);
    u_partial_kernel<<<dim3(C_EMB / 256, 8), 256, 0, stream>>>(W, qlast, part);
    u_reduce_kernel<<<C_EMB / 256, 256, 0, stream>>>(part, u);
    fused_att_v_kernel<<<T_TOK / 16 / 8, 256, 0, stream>>>(x, u, wv, att, vv);
    scan_kernel<<<1, 1024, 0, stream>>>(att, vv, alpha, beta, out);
}